// VectorQuantizer_69106023793030
// MI455X (gfx1250) — compile-verified
//
#include <hip/hip_runtime.h>

typedef __attribute__((ext_vector_type(2))) float v2f;
typedef __attribute__((ext_vector_type(8))) float v8f;

#define K_CODES 512
#define DIM 64
#define HW 4096                       // H*W = 64*64
#define N_TOTAL 131072                // 32*64*64
#define ROWS_PER_BLOCK 128
#define NBLOCKS (N_TOTAL / ROWS_PER_BLOCK)   // 1024
#define EROW 66                       // padded LDS row stride in floats (conflict-free b64)
#define SM_E (K_CODES * EROW)         // 33792 floats

// smem layout: sE[512*66] | hn[512] | red[256] | sIdx[128 ints]
#define SMEM_FLOATS (SM_E + K_CODES + 256 + 128)

__launch_bounds__(256, 1)
__global__ void vq_main(const float* __restrict__ lat,
                        const float* __restrict__ emb,
                        float* __restrict__ out,
                        float* __restrict__ partial) {
  extern __shared__ float smem[];
  float* sE   = smem;                       // 512 x 66
  float* hn   = smem + SM_E;                // 512 half-norms
  float* red  = hn + K_CODES;               // 256 reduction scratch
  int*   sIdx = (int*)(red + 256);          // 128 winning indices

  const int tid = threadIdx.x;

  // ---- stage 0: embedding -> LDS (padded), half-norms on the fly ----
  for (int r = tid; r < K_CODES; r += 256) {
    const float* src = emb + r * DIM;
    float* dst = sE + r * EROW;
    float s = 0.f;
#pragma unroll
    for (int d = 0; d < DIM; d += 2) {
      float x0 = src[d], x1 = src[d + 1];
      dst[d] = x0; dst[d + 1] = x1;         // 8B-aligned -> ds_store_b64
      s += x0 * x0 + x1 * x1;
    }
    hn[r] = 0.5f * s;
  }
  __syncthreads();

  // ---- stage 1: per-wave 16-row GEMM + fused argmin ----
  const int lane = tid & 31;
  const int wave = tid >> 5;
  const int l16  = lane & 15;
  const int dOff = (lane >> 4) << 1;        // 0 for lanes 0-15, 2 for lanes 16-31
  const int rowBase = blockIdx.x * ROWS_PER_BLOCK + wave * 16;
  const int bImg = rowBase >> 12;
  const int hw   = rowBase & (HW - 1);

  // A fragment: 16 rows x 64 d, fp32, per 16x16x4 layout
  const float* aPtr = lat + (size_t)bImg * (DIM * HW) + hw + l16;
  float aR[32];
#pragma unroll
  for (int c = 0; c < 16; ++c) {
    const int d0 = 4 * c + dOff;
    aR[2 * c]     = aPtr[(size_t)d0 * HW];
    aR[2 * c + 1] = aPtr[(size_t)(d0 + 1) * HW];
  }

  float bestV[8];
  int   bestI[8];
#pragma unroll
  for (int i = 0; i < 8; ++i) { bestV[i] = 3.4e38f; bestI[i] = 0; }

  for (int ct = 0; ct < 32; ++ct) {
    const int col = ct * 16 + l16;
    const float* eP = sE + col * EROW + dOff;
    v8f acc = {0.f, 0.f, 0.f, 0.f, 0.f, 0.f, 0.f, 0.f};
#pragma unroll
    for (int c = 0; c < 16; ++c) {
      v2f av; av.x = aR[2 * c];  av.y = aR[2 * c + 1];
      v2f bv; bv.x = eP[4 * c];  bv.y = eP[4 * c + 1];   // ds_load_b64, conflict-free
      acc = __builtin_amdgcn_wmma_f32_16x16x4_f32(false, av, false, bv,
                                                  (short)0, acc, false, false);
    }
    const float h = hn[col];
#pragma unroll
    for (int i = 0; i < 8; ++i) {
      const float v = h - acc[i];           // 0.5||e||^2 - x.e  (argmin-equivalent)
      if (v < bestV[i]) { bestV[i] = v; bestI[i] = col; }
    }
  }

  // butterfly argmin across the 16-lane half (rows 0-7 in lanes 0-15, 8-15 in 16-31)
#pragma unroll
  for (int off = 8; off > 0; off >>= 1) {
#pragma unroll
    for (int i = 0; i < 8; ++i) {
      const float ov = __shfl_xor(bestV[i], off, 32);
      const int   oi = __shfl_xor(bestI[i], off, 32);
      if (ov < bestV[i] || (ov == bestV[i] && oi < bestI[i])) {
        bestV[i] = ov; bestI[i] = oi;
      }
    }
  }
  if (l16 == 0) {
    const int mOff = (lane >> 4) * 8;
#pragma unroll
    for (int i = 0; i < 8; ++i)
      sIdx[wave * 16 + mOff + i] = bestI[i];
  }
  __syncthreads();

  // ---- stage 2: gather codes, permuted coalesced store, loss accumulate ----
  const int nBase = blockIdx.x * ROWS_PER_BLOCK;
  const int bO  = nBase >> 12;
  const int hwB = nBase & (HW - 1);
  const int j     = tid & 127;
  const int dHalf = tid >> 7;               // 0 or 1
  const float* eRow = sE + sIdx[j] * EROW;
  const size_t base = (size_t)bO * (DIM * HW) + hwB + j;
  float lsum = 0.f;
#pragma unroll
  for (int it = 0; it < 32; ++it) {
    const int d = dHalf + 2 * it;
    const float q = eRow[d];
    const size_t off = base + (size_t)d * HW;
    const float x = lat[off];
    out[off] = q;
    const float diff = q - x;
    lsum += diff * diff;
  }

  red[tid] = lsum;
  __syncthreads();
  for (int s = 128; s > 0; s >>= 1) {
    if (tid < s) red[tid] += red[tid + s];
    __syncthreads();
  }
  if (tid == 0) partial[blockIdx.x] = red[0];
}

__global__ void vq_finalize(const float* __restrict__ partial,
                            float* __restrict__ lossOut) {
  __shared__ float red[256];
  const int t = threadIdx.x;
  red[t] = partial[t] + partial[t + 256] + partial[t + 512] + partial[t + 768];
  __syncthreads();
  for (int s = 128; s > 0; s >>= 1) {
    if (t < s) red[t] += red[t + s];
    __syncthreads();
  }
  if (t == 0)
    lossOut[0] = red[0] * (1.25f / (float)(N_TOTAL * DIM));  // (beta+1)*MSE
}

extern "C" void kernel_launch(void* const* d_in, const int* in_sizes, int n_in,
                              void* d_out, int out_size, void* d_ws, size_t ws_size,
                              hipStream_t stream) {
  const float* lat = (const float*)d_in[0];   // [32,64,64,64] f32
  const float* emb = (const float*)d_in[1];   // [512,64] f32
  float* out = (float*)d_out;                 // [32,64,64,64] f32 ++ [1] loss
  float* partial = (float*)d_ws;              // 1024 block partials

  const int smemBytes = SMEM_FLOATS * 4;      // ~135.5 KB dynamic LDS (CDNA5: 320KB/WGP)
  hipFuncSetAttribute(reinterpret_cast<const void*>(vq_main),
                      hipFuncAttributeMaxDynamicSharedMemorySize, smemBytes);

  vq_main<<<NBLOCKS, 256, smemBytes, stream>>>(lat, emb, out, partial);
  vq_finalize<<<1, 256, 0, stream>>>(partial, out + (size_t)N_TOTAL * DIM);
}